// EveryStepLoss_28793460752554
// MI455X (gfx1250) — compile-verified
//
#include <hip/hip_runtime.h>
#include <hip/hip_bf16.h>

typedef __attribute__((ext_vector_type(2))) float v2f;
typedef __attribute__((ext_vector_type(8))) float v8f;

// ---------------------------------------------------------------------------
// wave32 butterfly reductions (width=32 explicitly: gfx1250 is wave32-only)
// ---------------------------------------------------------------------------
static __device__ __forceinline__ float wave_sum(float v) {
#pragma unroll
  for (int o = 16; o > 0; o >>= 1) v += __shfl_xor(v, o, 32);
  return v;
}
static __device__ __forceinline__ float wave_max(float v) {
#pragma unroll
  for (int o = 16; o > 0; o >>= 1) v = fmaxf(v, __shfl_xor(v, o, 32));
  return v;
}

// ---------------------------------------------------------------------------
// Kernel A: exclusive prefix sum of lengths -> segment starts (B is tiny)
// ---------------------------------------------------------------------------
__global__ void starts_kernel(const int* __restrict__ lens,
                              int* __restrict__ starts, int B) {
  if (blockIdx.x == 0 && threadIdx.x == 0) {
    int acc = 0;
    for (int b = 0; b < B; ++b) { starts[b] = acc; acc += lens[b]; }
  }
}

// ---------------------------------------------------------------------------
// Kernel B (hot, HBM-bound): per-token loss = logsumexp(row) - row[target].
// One wave per token; lane holds C/32 channels in registers (NC4 float4s),
// exact two-pass max -> sum(exp) entirely in registers: one HBM read total.
// ---------------------------------------------------------------------------
template <int NC4>
__global__ void token_loss_kernel(const float* __restrict__ outp,
                                  const int* __restrict__ tgt,
                                  float* __restrict__ loss,
                                  int T, int C) {
  const int lane = threadIdx.x & 31;
  const int wid  = (int)((blockIdx.x * blockDim.x + threadIdx.x) >> 5);
  const int nw   = (int)((gridDim.x * blockDim.x) >> 5);
  for (int t = wid; t < T; t += nw) {
    // CDNA5 prefetch path (global_prefetch_b8) for the next token this wave owns
    if (t + nw < T)
      __builtin_prefetch(outp + (size_t)(t + nw) * C + lane * 4, 0, 0);

    const float4* p = (const float4*)(outp + (size_t)t * C);
    float4 v[NC4];
#pragma unroll
    for (int i = 0; i < NC4; ++i) v[i] = p[lane + 32 * i];

    float m = -__builtin_inff();
#pragma unroll
    for (int i = 0; i < NC4; ++i)
      m = fmaxf(m, fmaxf(fmaxf(v[i].x, v[i].y), fmaxf(v[i].z, v[i].w)));
    m = wave_max(m);

    float s = 0.f;
#pragma unroll
    for (int i = 0; i < NC4; ++i)
      s += __expf(v[i].x - m) + __expf(v[i].y - m) +
           __expf(v[i].z - m) + __expf(v[i].w - m);
    s = wave_sum(s);

    if (lane == 0) {
      float xt = outp[(size_t)t * C + tgt[t]];
      loss[t] = m + __logf(s) - xt;
    }
  }
}

// Generic fallback (any C): per-lane online logsumexp, then lane merge.
__global__ void token_loss_kernel_gen(const float* __restrict__ outp,
                                      const int* __restrict__ tgt,
                                      float* __restrict__ loss,
                                      int T, int C) {
  const int lane = threadIdx.x & 31;
  const int wid  = (int)((blockIdx.x * blockDim.x + threadIdx.x) >> 5);
  const int nw   = (int)((gridDim.x * blockDim.x) >> 5);
  for (int t = wid; t < T; t += nw) {
    const float* row = outp + (size_t)t * C;
    float m = -__builtin_inff(), s = 0.f;
    for (int c = lane; c < C; c += 32) {
      float x  = row[c];
      float nm = fmaxf(m, x);
      s = s * __expf(m - nm) + __expf(x - nm);
      m = nm;
    }
#pragma unroll
    for (int o = 16; o > 0; o >>= 1) {
      float om = __shfl_xor(m, o, 32);
      float os = __shfl_xor(s, o, 32);
      float nm = fmaxf(m, om);
      s = s * __expf(m - nm) + os * __expf(om - nm);
      m = nm;
    }
    if (lane == 0) loss[t] = m + __logf(s) - row[tgt[t]];
  }
}

// ---------------------------------------------------------------------------
// Kernel C: per-segment  partial[b] = (sum_p loss[p]*e[p]) / (sum_p e[p]),
// e(p) = exp(2g*p/denom - 2g)  (analytic, no memory traffic).
// The weighted dot runs on V_WMMA_F32_16X16X4_F32: A(16x4)=loss chunk,
// B(4x16)=e chunk, diag of the f32 accumulator = 16 partial dots per issue.
// One fully-active wave per segment (EXEC all ones for WMMA); tails masked
// branchlessly with zeros.
// ---------------------------------------------------------------------------
__global__ void seg_dot_kernel(const float* __restrict__ loss,
                               const int* __restrict__ lens,
                               const int* __restrict__ starts,
                               const float* __restrict__ gamma_p,
                               float* __restrict__ partial, int B) {
  const int b    = blockIdx.x;
  const int lane = threadIdx.x & 31;
  const int L     = lens[b];
  const int start = starts[b];
  const float g   = *gamma_p;
  const float denom = fmaxf((float)(L - 1), 1.0f);
  const float scale = 2.0f * g / denom;
  const float c0    = -2.0f * g;

  // sum of weights (exact, fixed lane order)
  float ssum = 0.f;
  for (int p = lane; p < L; p += 32) ssum += __expf(c0 + scale * (float)p);
  ssum = wave_sum(ssum);

  // WMMA f32 accumulation of loss . e over 64-token chunks
  const int half = lane >> 4;   // 0: lanes 0-15, 1: lanes 16-31
  const int mm   = lane & 15;   // row / col index within tile
  v8f d = {0.f, 0.f, 0.f, 0.f, 0.f, 0.f, 0.f, 0.f};
  const int nchunks = (L + 63) >> 6;
  for (int ch = 0; ch < nchunks; ++ch) {
    const int q = ch << 6;
    // A(16x4) ISA layout: lane<16 holds (M=mm,K=0..1), lane>=16 (M=mm,K=2..3)
    const int pa = q + 4 * mm + 2 * half;
    v2f a;
    a.x = (pa     < L) ? loss[start + pa]     : 0.f;
    a.y = (pa + 1 < L) ? loss[start + pa + 1] : 0.f;
    // B(4x16) layout: VGPR0 = rows {K=0 (lanes<16), K=1}, VGPR1 = rows {K=2, K=3}
    const int pb0 = q + 4 * mm + half;
    const int pb1 = pb0 + 2;
    v2f bb;
    bb.x = (pb0 < L) ? __expf(c0 + scale * (float)pb0) : 0.f;
    bb.y = (pb1 < L) ? __expf(c0 + scale * (float)pb1) : 0.f;
    // D[m][j] += sum_k A[m][k]*B[k][j]; diagonal m==j = per-group-of-4 dots
    d = __builtin_amdgcn_wmma_f32_16x16x4_f32(false, a, false, bb,
                                              (short)0, d, false, false);
  }
  // Extract diagonal: D(m,m) m<8 -> vgpr m, lane m ; m>=8 -> vgpr m-8, lane m+16
  float acc = 0.f;
#pragma unroll
  for (int r = 0; r < 8; ++r)
    acc += ((lane == r) || (lane == r + 24)) ? d[r] : 0.f;
  acc = wave_sum(acc);

  if (lane == 0) partial[b] = (L > 0) ? (acc / ssum) : 0.f;
}

// ---------------------------------------------------------------------------
// Kernel D: final deterministic combine
// ---------------------------------------------------------------------------
__global__ void final_kernel(const float* __restrict__ partial,
                             float* __restrict__ out, int B) {
  const int lane = threadIdx.x & 31;
  float s = 0.f;
  for (int b = lane; b < B; b += 32) s += partial[b];
  s = wave_sum(s);
  if (lane == 0) out[0] = s / (float)B;
}

// ---------------------------------------------------------------------------
extern "C" void kernel_launch(void* const* d_in, const int* in_sizes, int n_in,
                              void* d_out, int out_size, void* d_ws, size_t ws_size,
                              hipStream_t stream) {
  const float* outp    = (const float*)d_in[0];
  const int*   tgt     = (const int*)d_in[1];
  const int*   lens    = (const int*)d_in[2];
  const float* gamma_p = (const float*)d_in[3];

  const int TC = in_sizes[0];
  const int T  = in_sizes[1];
  const int B  = in_sizes[2];
  const int C  = TC / T;

  char* ws = (char*)d_ws;
  int* starts = (int*)ws;
  size_t off = ((size_t)B * sizeof(int) + 255) & ~(size_t)255;
  float* partial = (float*)(ws + off);
  off += (((size_t)B * sizeof(float) + 255) & ~(size_t)255);
  float* loss = (float*)(ws + off);

  starts_kernel<<<dim3(1), dim3(32), 0, stream>>>(lens, starts, B);

  int blocks = (T + 7) / 8;            // 8 waves per 256-thread block
  if (blocks > 8192) blocks = 8192;
  if (blocks < 1) blocks = 1;
  switch (C) {
    case 256:  token_loss_kernel<2><<<blocks, 256, 0, stream>>>(outp, tgt, loss, T, C);  break;
    case 512:  token_loss_kernel<4><<<blocks, 256, 0, stream>>>(outp, tgt, loss, T, C);  break;
    case 1024: token_loss_kernel<8><<<blocks, 256, 0, stream>>>(outp, tgt, loss, T, C);  break;
    case 2048: token_loss_kernel<16><<<blocks, 256, 0, stream>>>(outp, tgt, loss, T, C); break;
    default:   token_loss_kernel_gen<<<blocks, 256, 0, stream>>>(outp, tgt, loss, T, C); break;
  }

  seg_dot_kernel<<<dim3(B), dim3(32), 0, stream>>>(loss, lens, starts, gamma_p, partial, B);
  final_kernel<<<dim3(1), dim3(32), 0, stream>>>(partial, (float*)d_out, B);
}